// ApproxNDCGLoss_5609227288644
// MI455X (gfx1250) — compile-verified
//
#include <hip/hip_runtime.h>
#include <hip/hip_bf16.h>

typedef __attribute__((ext_vector_type(2))) float v2f;
typedef __attribute__((ext_vector_type(8))) float v8f;

#define WAVES_PER_BLOCK 8   // 256 threads = 8 wave32

// ---------------------------------------------------------------------------
// Kernel 1: approx_rank[j] = 1 + sum_i sigmoid(s[j] - s[i])      (TEMPERATURE=1)
//
// Identity: sum_i sigmoid(s_j - s_i) = n/2 + (1/2) * sum_i tanh((s_j-s_i)/2).
// LDS holds h = s/2, so the hot loop per element is just v_sub_f32 +
// v_tanh_f32 (gfx1250 TRANS op), feeding a 4x16 f32 B-tile that is reduced
// with D = ones(16x4) x B + C on V_WMMA_F32_16X16X4_F32. Every row of D
// carries the same column sum; lanes 0..15 read it from VGPR0 (M=0,N=lane).
// Full f32 accumulation, zero cross-lane shuffles, 1 TRANS op per element.
// ---------------------------------------------------------------------------
__global__ void rank_wmma_kernel(const float* __restrict__ s,
                                 float* __restrict__ rank, int n) {
  extern __shared__ float sh[];                 // n floats: h = s/2
  const int tid = threadIdx.x;
  for (int k = tid; k < n; k += blockDim.x) sh[k] = 0.5f * s[k];
  __syncthreads();

  const int wave  = tid >> 5;
  const int lane  = tid & 31;
  const int jbase = (blockIdx.x * WAVES_PER_BLOCK + wave) * 16;
  if (jbase >= n) return;                       // wave-uniform exit

  const int col  = lane & 15;                   // N-column within tile
  const int half = lane >> 4;                   // K half (K0/K1 vs K2/K3)
  const int jcol = jbase + col;
  const float hj = sh[(jcol < n) ? jcol : (n - 1)];

  v2f a; a.x = 1.0f; a.y = 1.0f;                // A = ones(16x4)
  v8f c = {};                                   // f32 accumulator tile
  const int n4 = n & ~3;
  for (int i = 0; i < n4; i += 4) {
    const int i0 = i + half * 2;                // 8B-aligned pair -> ds_load_b64
    v2f b;
    b.x = __builtin_amdgcn_tanhf(hj - sh[i0]);      // v_tanh_f32 (TRANS)
    b.y = __builtin_amdgcn_tanhf(hj - sh[i0 + 1]);
    c = __builtin_amdgcn_wmma_f32_16x16x4_f32(
        /*neg_a=*/false, a, /*neg_b=*/false, b,
        /*c_mod=*/(short)0, c, /*reuse_a=*/false, /*reuse_b=*/false);
  }

  if (lane < 16 && jcol < n) {
    float tanh_sum = c[0];                      // D row 0, column = lane
    for (int i = n4; i < n; ++i)                // i-tail (empty when 4 | n)
      tanh_sum += __builtin_amdgcn_tanhf(hj - sh[i]);
    rank[jcol] = 0.5f * tanh_sum + 0.5f * (float)n + 1.0f;
  }
}

// ---------------------------------------------------------------------------
// Kernel 2: per-element DCG / IDCG / sum(y) contributions + per-block partials.
// IDCG avoids an explicit sort: the descending-sort position of y[j] is
// #{y_i > y_j} + #{y_i == y_j, i < j} (ties contribute identically to IDCG).
// ---------------------------------------------------------------------------
__global__ void dcg_kernel(const float* __restrict__ y,
                           const float* __restrict__ rank,
                           float* __restrict__ pD, float* __restrict__ pI,
                           float* __restrict__ pS, int n) {
  extern __shared__ float shy[];                // n floats: whole targets vector
  __shared__ float rD[256], rI[256], rS[256];
  const int tid = threadIdx.x;
  for (int k = tid; k < n; k += blockDim.x) shy[k] = y[k];
  __syncthreads();

  const int j = blockIdx.x * blockDim.x + tid;
  const bool valid = (j < n);
  const float yj = valid ? shy[j] : 0.0f;

  int cnt = 0;
#pragma unroll 4
  for (int i = 0; i < n; ++i) {
    const float yi = shy[i];
    cnt += (yi > yj) || ((yi == yj) && (i < j));
  }

  float dc = 0.0f, ic = 0.0f, sc = 0.0f;
  if (valid) {
    const float r = rank[j];
    dc = yj / __log2f(r + 1.0f);                // y * discount(approx rank)
    ic = yj / __log2f((float)(cnt + 2));        // y * ideal discount(position)
    sc = yj;
  }

  rD[tid] = dc; rI[tid] = ic; rS[tid] = sc;
  __syncthreads();
  for (int off = 128; off > 0; off >>= 1) {     // deterministic tree reduce
    if (tid < off) {
      rD[tid] += rD[tid + off];
      rI[tid] += rI[tid + off];
      rS[tid] += rS[tid + off];
    }
    __syncthreads();
  }
  if (tid == 0) {
    pD[blockIdx.x] = rD[0];
    pI[blockIdx.x] = rI[0];
    pS[blockIdx.x] = rS[0];
  }
}

// ---------------------------------------------------------------------------
// Kernel 3: final deterministic reduction + loss.
// ---------------------------------------------------------------------------
__global__ void final_kernel(const float* __restrict__ pD,
                             const float* __restrict__ pI,
                             const float* __restrict__ pS,
                             float* __restrict__ out, int m) {
  __shared__ float sD[128], sI[128], sS[128];
  const int t = threadIdx.x;
  float d = 0.0f, i_ = 0.0f, s_ = 0.0f;
  for (int k = t; k < m; k += 128) { d += pD[k]; i_ += pI[k]; s_ += pS[k]; }
  sD[t] = d; sI[t] = i_; sS[t] = s_;
  __syncthreads();
  for (int off = 64; off > 0; off >>= 1) {
    if (t < off) { sD[t] += sD[t + off]; sI[t] += sI[t + off]; sS[t] += sS[t + off]; }
    __syncthreads();
  }
  if (t == 0) {
    const float ndcg = sD[0] / (sI[0] + 1e-8f);
    const float loss = 1.0f - ndcg;
    out[0] = (sS[0] < 1.0f) ? 0.0f : loss;      // early-return guard
  }
}

extern "C" void kernel_launch(void* const* d_in, const int* in_sizes, int n_in,
                              void* d_out, int out_size, void* d_ws, size_t ws_size,
                              hipStream_t stream) {
  const float* s = (const float*)d_in[0];       // logits
  const float* y = (const float*)d_in[1];       // targets
  const int n = in_sizes[0];

  // Workspace layout: rank[n] | pD[blocks2] | pI[blocks2] | pS[blocks2]
  float* rank = (float*)d_ws;
  const int blocks2 = (n + 255) / 256;
  float* pD = rank + n;
  float* pI = pD + blocks2;
  float* pS = pI + blocks2;

  const size_t lds_bytes = (size_t)n * sizeof(float);   // 80 KB for n=20000
  const int nwaves  = (n + 15) / 16;
  const int blocks1 = (nwaves + WAVES_PER_BLOCK - 1) / WAVES_PER_BLOCK;

  rank_wmma_kernel<<<blocks1, WAVES_PER_BLOCK * 32, lds_bytes, stream>>>(s, rank, n);
  dcg_kernel<<<blocks2, 256, lds_bytes, stream>>>(y, rank, pD, pI, pS, n);
  final_kernel<<<1, 128, 0, stream>>>(pD, pI, pS, (float*)d_out, blocks2);
}